// GCN_12859132084330
// MI455X (gfx1250) — compile-verified
//
#include <hip/hip_runtime.h>
#include <hip/hip_bf16.h>
#include <math.h>

typedef float v2f __attribute__((ext_vector_type(2)));
typedef float v8f __attribute__((ext_vector_type(8)));

#define N_FEAT 512
#define HIDDEN 64
#define N_CLASSES 40
#define GEMM_WAVES 5                 // 5 waves/block: 80 rows/block, 100000/80 = 1250 blocks exact
#define GEMM_THREADS (GEMM_WAVES * 32)

#if defined(__has_builtin)
#if __has_builtin(__builtin_amdgcn_global_load_async_to_lds_b128)
#define ASYNC_LDS 1
#endif
#endif
#ifndef ASYNC_LDS
#define ASYNC_LDS 0
#endif

#if ASYNC_LDS
// Matches the builtin's parameter types as reported by clang:
//   param0: __attribute__((vector_size(16))) int __device__ *   (addrspace(1))
//   param1: LDS-side pointer (addrspace(3)), then imm offset, imm cpol
typedef int v4i_gcc __attribute__((vector_size(16)));
typedef __attribute__((address_space(1))) v4i_gcc* as1_v4i_p;
typedef __attribute__((address_space(3))) v4i_gcc* as3_v4i_p;
#endif

// ---------------------------------------------------------------------------
// deg[dst] += 1 per edge (float atomics; deg array is tiny and L2-resident)
// ---------------------------------------------------------------------------
__global__ void gcn_degree_kernel(const int* __restrict__ dst, float* __restrict__ deg, int E) {
    int e = blockIdx.x * blockDim.x + threadIdx.x;
    if (e < E) atomicAdd(deg + dst[e], 1.0f);
}

// dis[i] = rsqrt(deg[i] + 1)   (self-loop) -- in place
__global__ void gcn_rsqrt_kernel(float* __restrict__ deg, int N) {
    int i = blockIdx.x * blockDim.x + threadIdx.x;
    if (i < N) deg[i] = rsqrtf(deg[i] + 1.0f);
}

// ---------------------------------------------------------------------------
// fp32 WMMA GEMM, B staged in LDS: block = 5 waves, each wave owns a 16-row
// tile x all 64 cols; B [K x 64] (<=128 KB, fits the 320 KB WGP LDS) is loaded
// once per block and read via ds_load in the K-loop.
// A-matrix 16x4 f32 layout: lanes 0-15 rows; v0 = K=0 (lanes lo)/K=2 (hi).
// B-matrix 4x16 f32 layout (mirror): v0 = row kb, v1 = row kb+1, lanes = cols.
// ---------------------------------------------------------------------------
template <int K>
__global__ __launch_bounds__(GEMM_THREADS) void gcn_block_gemm_n64_kernel(
        const float* __restrict__ A, const float* __restrict__ B, float* __restrict__ C) {
    extern __shared__ float shB[];   // K * 64 floats

    // ---- cooperative stage of B into LDS (async-to-LDS when available) ----
    const int vec_total = K * 16;    // number of float4 chunks
    for (int i = threadIdx.x; i < vec_total; i += GEMM_THREADS) {
#if ASYNC_LDS
        __builtin_amdgcn_global_load_async_to_lds_b128(
            (as1_v4i_p)(B + 4 * i), (as3_v4i_p)(shB + 4 * i), 0, 0);
#else
        *(float4*)(shB + 4 * i) = *(const float4*)(B + 4 * i);
#endif
    }
#if ASYNC_LDS
#if __has_builtin(__builtin_amdgcn_s_wait_asynccnt)
    __builtin_amdgcn_s_wait_asynccnt(0);
#else
    asm volatile("s_wait_asynccnt 0" ::: "memory");
#endif
#endif
    __syncthreads();

    const int lane  = threadIdx.x & 31;
    const int wid   = threadIdx.x >> 5;                 // 0..4
    const int m     = lane & 15;
    const int khalf = (lane >> 4) << 1;                 // 0 or 2
    const int base  = blockIdx.x * (GEMM_WAVES * 16) + wid * 16;

    v8f c0 = {}, c1 = {}, c2 = {}, c3 = {};
    const float* __restrict__ arow = A + (size_t)(base + m) * K;

    for (int k0 = 0; k0 < K; k0 += 4) {
        if ((k0 & 63) == 0 && k0 + 64 < K)
            __builtin_prefetch(arow + k0 + 64, 0, 1);   // uniform cond -> global_prefetch
        const int kb = k0 + khalf;
        v2f a;
        a.x = arow[kb];
        a.y = arow[kb + 1];
        const float* __restrict__ br0 = shB + kb * HIDDEN;
        const float* __restrict__ br1 = br0 + HIDDEN;

        v2f b;
        b.x = br0[m];       b.y = br1[m];
        c0 = __builtin_amdgcn_wmma_f32_16x16x4_f32(false, a, false, b, (short)0, c0, false, false);
        b.x = br0[m + 16];  b.y = br1[m + 16];
        c1 = __builtin_amdgcn_wmma_f32_16x16x4_f32(false, a, false, b, (short)0, c1, false, false);
        b.x = br0[m + 32];  b.y = br1[m + 32];
        c2 = __builtin_amdgcn_wmma_f32_16x16x4_f32(false, a, false, b, (short)0, c2, false, false);
        b.x = br0[m + 48];  b.y = br1[m + 48];
        c3 = __builtin_amdgcn_wmma_f32_16x16x4_f32(false, a, false, b, (short)0, c3, false, false);
    }

    // C/D layout: VGPR j holds row M=j (lanes 0-15) and M=j+8 (lanes 16-31), lane = N.
    const int mh = (lane >> 4) * 8;
#pragma unroll
    for (int j = 0; j < 8; ++j) {
        float* __restrict__ cp = C + (size_t)(base + j + mh) * HIDDEN;
        cp[m]      = c0[j];
        cp[m + 16] = c1[j];
        cp[m + 32] = c2[j];
        cp[m + 48] = c3[j];
    }
}

// ---------------------------------------------------------------------------
// Edge aggregation: agg[dst] += h[src] * dis[src]*dis[dst]
// 16 threads per edge, float4 per thread. h (25.6 MB) is L2-resident -> the
// random gather and the f32 atomics both hit L2.
// ---------------------------------------------------------------------------
__global__ void gcn_aggregate_kernel(const int* __restrict__ src, const int* __restrict__ dst,
                                     const float* __restrict__ dis, const float* __restrict__ h,
                                     float* __restrict__ agg, int E) {
    size_t tid = (size_t)blockIdx.x * blockDim.x + threadIdx.x;
    int e = (int)(tid >> 4);
    if (e >= E) return;
    int f4 = (int)(tid & 15) << 2;
    int s = src[e], d = dst[e];
    float w = dis[s] * dis[d];
    const float4 v = *(const float4*)(h + (size_t)s * HIDDEN + f4);
    float* __restrict__ ap = agg + (size_t)d * HIDDEN + f4;
    atomicAdd(ap + 0, v.x * w);
    atomicAdd(ap + 1, v.y * w);
    atomicAdd(ap + 2, v.z * w);
    atomicAdd(ap + 3, v.w * w);
}

// out[i][f] = relu(agg[i][f] + hW[i][f]*dis[i]^2 + bias[f])
__global__ void gcn_combine_relu_kernel(const float* __restrict__ agg, const float* __restrict__ hW,
                                        const float* __restrict__ dis, const float* __restrict__ bias,
                                        float* __restrict__ out, int N) {
    size_t tid = (size_t)blockIdx.x * blockDim.x + threadIdx.x;
    int i = (int)(tid >> 4);
    if (i >= N) return;
    int f4 = (int)(tid & 15) << 2;
    float sl = dis[i];
    sl *= sl;
    const float4 a = *(const float4*)(agg + (tid << 2));
    const float4 hv = *(const float4*)(hW + (tid << 2));
    const float4 b = *(const float4*)(bias + f4);
    float4 r;
    r.x = fmaxf(fmaf(hv.x, sl, a.x) + b.x, 0.0f);
    r.y = fmaxf(fmaf(hv.y, sl, a.y) + b.y, 0.0f);
    r.z = fmaxf(fmaf(hv.z, sl, a.z) + b.z, 0.0f);
    r.w = fmaxf(fmaf(hv.w, sl, a.w) + b.w, 0.0f);
    *(float4*)(out + (tid << 2)) = r;
}

// ---------------------------------------------------------------------------
// Output GEMM: logits[N x 40] = A[N x 64] @ W_out[64 x 40] + b_out.
// N=40 padded to 3 WMMA N-tiles; out-of-range B columns selected to 0 with
// cndmask (no divergent branch before WMMA, EXEC stays all-ones).
// ---------------------------------------------------------------------------
__global__ __launch_bounds__(32) void gcn_wave_gemm_out_kernel(
        const float* __restrict__ A, const float* __restrict__ B,
        const float* __restrict__ bias, float* __restrict__ L) {
    const int lane  = threadIdx.x;
    const int m     = lane & 15;
    const int khalf = (lane >> 4) << 1;
    const int base  = blockIdx.x * 16;

    v8f c0 = {}, c1 = {}, c2 = {};
    const float* __restrict__ arow = A + (size_t)(base + m) * HIDDEN;

    const int n2    = m + 32;
    const bool v2ok = (n2 < N_CLASSES);
    const int n2c   = v2ok ? n2 : (N_CLASSES - 1);     // clamped addr, value selected to 0

    for (int k0 = 0; k0 < HIDDEN; k0 += 4) {
        const int kb = k0 + khalf;
        v2f a;
        a.x = arow[kb];
        a.y = arow[kb + 1];
        const float* __restrict__ br0 = B + (size_t)kb * N_CLASSES;
        const float* __restrict__ br1 = br0 + N_CLASSES;

        v2f b;
        b.x = br0[m];       b.y = br1[m];
        c0 = __builtin_amdgcn_wmma_f32_16x16x4_f32(false, a, false, b, (short)0, c0, false, false);
        b.x = br0[m + 16];  b.y = br1[m + 16];
        c1 = __builtin_amdgcn_wmma_f32_16x16x4_f32(false, a, false, b, (short)0, c1, false, false);
        float t0 = br0[n2c], t1 = br1[n2c];
        b.x = v2ok ? t0 : 0.0f;
        b.y = v2ok ? t1 : 0.0f;
        c2 = __builtin_amdgcn_wmma_f32_16x16x4_f32(false, a, false, b, (short)0, c2, false, false);
    }

    const int mh = (lane >> 4) * 8;
    const float b0 = bias[m], b1v = bias[m + 16];
    const float b2v = v2ok ? bias[n2] : 0.0f;
#pragma unroll
    for (int j = 0; j < 8; ++j) {
        float* __restrict__ lp = L + (size_t)(base + j + mh) * N_CLASSES;
        lp[m]      = c0[j] + b0;
        lp[m + 16] = c1[j] + b1v;
        if (v2ok) lp[n2] = c2[j] + b2v;
    }
}

// Row softmax over 40 classes, one thread per node (16 MB traffic, trivial).
__global__ void gcn_softmax_kernel(const float* __restrict__ logits, float* __restrict__ out, int N) {
    int i = blockIdx.x * blockDim.x + threadIdx.x;
    if (i >= N) return;
    const float* __restrict__ lp = logits + (size_t)i * N_CLASSES;
    float mx = -INFINITY;
#pragma unroll
    for (int c = 0; c < N_CLASSES; ++c) mx = fmaxf(mx, lp[c]);
    float ex[N_CLASSES];
    float sum = 0.0f;
#pragma unroll
    for (int c = 0; c < N_CLASSES; ++c) { ex[c] = __expf(lp[c] - mx); sum += ex[c]; }
    const float inv = 1.0f / sum;
    float* __restrict__ op = out + (size_t)i * N_CLASSES;
#pragma unroll
    for (int c = 0; c < N_CLASSES; ++c) op[c] = ex[c] * inv;
}

// ---------------------------------------------------------------------------
extern "C" void kernel_launch(void* const* d_in, const int* in_sizes, int n_in,
                              void* d_out, int out_size, void* d_ws, size_t ws_size,
                              hipStream_t stream) {
    const float* x     = (const float*)d_in[0];
    const int*   ei    = (const int*)d_in[1];
    const float* W1    = (const float*)d_in[2];
    const float* b1    = (const float*)d_in[3];
    const float* W2    = (const float*)d_in[4];
    const float* b2    = (const float*)d_in[5];
    const float* W_out = (const float*)d_in[6];
    const float* b_out = (const float*)d_in[7];
    float* out = (float*)d_out;

    const int N = in_sizes[0] / N_FEAT;   // 100000 (divisible by 80 and 16)
    const int E = in_sizes[1] / 2;        // 1600000
    const int* src = ei;
    const int* dst = ei + E;

    // Workspace carve-out (~94 MB)
    char* w = (char*)d_ws;
    size_t o = 0;
    auto carve = [&](size_t bytes) { void* p = w + o; o += (bytes + 255) & ~(size_t)255; return p; };
    float* dis    = (float*)carve((size_t)N * 4);            // degree -> rsqrt, in place
    float* hW     = (float*)carve((size_t)N * HIDDEN * 4);   // GEMM output of current layer
    float* hAct   = (float*)carve((size_t)N * HIDDEN * 4);   // activations (layer input)
    float* agg    = (float*)carve((size_t)N * HIDDEN * 4);   // scatter-add target (reused)
    float* logits = (float*)carve((size_t)N * N_CLASSES * 4);

    const int elemBlocks  = (int)(((size_t)N * 16 + 255) / 256);
    const int edgeBlocksF = (int)(((size_t)E * 16 + 255) / 256);
    const int gemmBlocks  = N / (GEMM_WAVES * 16);           // 1250

    // degree + norm
    (void)hipMemsetAsync(dis, 0, (size_t)N * 4, stream);
    (void)hipMemsetAsync(agg, 0, (size_t)N * HIDDEN * 4, stream);
    gcn_degree_kernel<<<(E + 255) / 256, 256, 0, stream>>>(dst, dis, E);
    gcn_rsqrt_kernel<<<(N + 255) / 256, 256, 0, stream>>>(dis, N);

    // layer 1 (B stage: 512*64*4 = 128 KB of dynamic LDS)
    gcn_block_gemm_n64_kernel<N_FEAT>
        <<<gemmBlocks, GEMM_THREADS, N_FEAT * HIDDEN * 4, stream>>>(x, W1, hW);
    gcn_aggregate_kernel<<<edgeBlocksF, 256, 0, stream>>>(src, dst, dis, hW, agg, E);
    gcn_combine_relu_kernel<<<elemBlocks, 256, 0, stream>>>(agg, hW, dis, b1, hAct, N);

    // layer 2 (B stage: 64*64*4 = 16 KB of dynamic LDS)
    (void)hipMemsetAsync(agg, 0, (size_t)N * HIDDEN * 4, stream);
    gcn_block_gemm_n64_kernel<HIDDEN>
        <<<gemmBlocks, GEMM_THREADS, HIDDEN * HIDDEN * 4, stream>>>(hAct, W2, hW);
    gcn_aggregate_kernel<<<edgeBlocksF, 256, 0, stream>>>(src, dst, dis, hW, agg, E);
    gcn_combine_relu_kernel<<<elemBlocks, 256, 0, stream>>>(agg, hW, dis, b2, hAct, N);

    // output head + softmax
    gcn_wave_gemm_out_kernel<<<N / 16, 32, 0, stream>>>(hAct, W_out, b_out, logits);
    gcn_softmax_kernel<<<(N + 255) / 256, 256, 0, stream>>>(logits, out, N);
}